// EdgeLLMAttention_82738249990458
// MI455X (gfx1250) — compile-verified
//
#include <hip/hip_runtime.h>
#include <hip/hip_bf16.h>

// ---- problem constants (match reference) ----
#define BB   2
#define SS   512
#define HID  2048
#define HQ   16
#define HKV  4
#define DD   128
#define GG   (HQ / HKV)
#define MAXP 4096
#define NEGV (-1.0e30f)

typedef float v2f __attribute__((ext_vector_type(2)));
typedef float v8f __attribute__((ext_vector_type(8)));
typedef int   v4i __attribute__((vector_size(16)));

// pointer types matching __builtin_amdgcn_global_load_async_to_lds_b128:
//   arg0: int4* in global (AS1 / "__device__"), arg1: int4* in LDS (AS3)
typedef __attribute__((address_space(1))) v4i* as1_v4i_ptr;
typedef __attribute__((address_space(3))) v4i* as3_v4i_ptr;

#define HAS_ASYNC_LDS __has_builtin(__builtin_amdgcn_global_load_async_to_lds_b128)

#if HAS_ASYNC_LDS
__device__ __forceinline__ void async_copy16(const float* g, float* l) {
  __builtin_amdgcn_global_load_async_to_lds_b128((as1_v4i_ptr)g, (as3_v4i_ptr)l, 0, 0);
}
#endif

#if __has_builtin(__builtin_amdgcn_s_wait_asynccnt)
#define WAIT_ASYNC(n) __builtin_amdgcn_s_wait_asynccnt(n)
#else
#define WAIT_ASYNC(n) asm volatile("s_wait_asynccnt %0" ::"i"(n) : "memory")
#endif

// =====================================================================
// Row-major f32 GEMM: C[M,N] = A[M,K] @ B[K,N]
// 128x64 block tile, BK=16, 256 threads = 8 waves, each wave 32x32
// (4 accumulators) via V_WMMA_F32_16X16X4_F32.
// M multiple of 128; N multiple of 64; K multiple of 16.
// Double-buffered GLOBAL_LOAD_ASYNC_TO_LDS staging when available.
// =====================================================================
__global__ __launch_bounds__(256) void gemm_f32_wmma(
    const float* __restrict__ A, const float* __restrict__ Bm,
    float* __restrict__ C, int M, int N, int K) {
  __shared__ float As[2][128][20];   // row stride 20: 16B aligned + conflict-free
  __shared__ float Bs[2][16][64];

  const int tid  = threadIdx.x;
  const int lane = tid & 31;
  const int wid  = tid >> 5;
  const int wm   = wid & 3;          // 4 M positions of 32 rows
  const int wn   = wid >> 2;         // 2 N positions of 32 cols
  const int lm   = lane & 15;
  const int hi   = lane >> 4;

  const int m0 = blockIdx.y * 128;
  const int n0 = blockIdx.x * 64;

  // cooperative tile load mapping
  const int arow = tid >> 2,  asc = (tid & 3)  * 4;   // A: 128x16 (two rows/thread)
  const int brow = tid >> 4,  bsc = (tid & 15) * 4;   // B: 16x64  (one seg/thread)

  v8f c00 = {}, c01 = {}, c10 = {}, c11 = {};
  const int T = K >> 4;

  auto compute = [&](int st) {
#pragma unroll
    for (int ki = 0; ki < 16; ki += 4) {
      v2f a0, a1, b0, b1;
      a0.x = As[st][wm * 32 + lm][ki + hi * 2 + 0];
      a0.y = As[st][wm * 32 + lm][ki + hi * 2 + 1];
      a1.x = As[st][wm * 32 + 16 + lm][ki + hi * 2 + 0];
      a1.y = As[st][wm * 32 + 16 + lm][ki + hi * 2 + 1];
      b0.x = Bs[st][ki + hi * 2 + 0][wn * 32 + lm];
      b0.y = Bs[st][ki + hi * 2 + 1][wn * 32 + lm];
      b1.x = Bs[st][ki + hi * 2 + 0][wn * 32 + 16 + lm];
      b1.y = Bs[st][ki + hi * 2 + 1][wn * 32 + 16 + lm];
      c00 = __builtin_amdgcn_wmma_f32_16x16x4_f32(false, a0, false, b0, (short)0, c00, false, false);
      c01 = __builtin_amdgcn_wmma_f32_16x16x4_f32(false, a0, false, b1, (short)0, c01, false, false);
      c10 = __builtin_amdgcn_wmma_f32_16x16x4_f32(false, a1, false, b0, (short)0, c10, false, false);
      c11 = __builtin_amdgcn_wmma_f32_16x16x4_f32(false, a1, false, b1, (short)0, c11, false, false);
    }
  };

#if HAS_ASYNC_LDS
  auto stage = [&](int st, int kk) {
    async_copy16(A  + (size_t)(m0 + arow)      * K + kk + asc, &As[st][arow][asc]);
    async_copy16(A  + (size_t)(m0 + 64 + arow) * K + kk + asc, &As[st][64 + arow][asc]);
    async_copy16(Bm + (size_t)(kk + brow)      * N + n0 + bsc, &Bs[st][brow][bsc]);
  };
  stage(0, 0);
  for (int t = 0; t < T; ++t) {
    const int cur = t & 1;
    if (t + 1 < T) {                 // prefetch next tile into other buffer
      stage(cur ^ 1, (t + 1) << 4);
      WAIT_ASYNC(3);                 // own stage-`cur` loads retired (in-order)
    } else {
      WAIT_ASYNC(0);
    }
    __syncthreads();                 // everyone's stage-`cur` data in LDS
    compute(cur);
    __syncthreads();                 // `cur` consumed before overwrite at t+2
  }
#else
  for (int t = 0; t < T; ++t) {
    const int kk = t << 4;
    const float4 av0 = *(const float4*)(A  + (size_t)(m0 + arow)      * K + kk + asc);
    const float4 av1 = *(const float4*)(A  + (size_t)(m0 + 64 + arow) * K + kk + asc);
    const float4 bv  = *(const float4*)(Bm + (size_t)(kk + brow)      * N + n0 + bsc);
    __syncthreads();                 // previous tile fully consumed
    *(float4*)(&As[0][arow][asc])      = av0;
    *(float4*)(&As[0][64 + arow][asc]) = av1;
    *(float4*)(&Bs[0][brow][bsc])      = bv;
    __syncthreads();
    compute(0);
  }
#endif

#pragma unroll
  for (int r = 0; r < 8; ++r) {
    const int row = m0 + wm * 32 + r + hi * 8;
    const int col = n0 + wn * 32 + lm;
    C[(size_t)row * N + col]             = c00[r];
    C[(size_t)row * N + col + 16]        = c01[r];
    C[(size_t)(row + 16) * N + col]      = c10[r];
    C[(size_t)(row + 16) * N + col + 16] = c11[r];
  }
}

// =====================================================================
// RMSNorm + RoPE + KV-cache scatter. One wave per (b, s, slot):
// slot 0..15 = Q head (norm+rope, in-place in qbuf)
// slot 16..19 = K head (norm+rope, scatter into present[ b,0,h,start+s ])
// slot 20..23 = V head (copy into present[ b,1,h,start+s ])
// =====================================================================
__global__ __launch_bounds__(32) void normrope_scatter(
    const float* __restrict__ qin, const float* __restrict__ kin,
    const float* __restrict__ vin, const float* __restrict__ rope,
    const float* __restrict__ qw,  const float* __restrict__ kw,
    float* __restrict__ qout, float* __restrict__ present,
    const int* __restrict__ start_idx) {
  int blk = blockIdx.x;
  const int slot = blk % 24; blk /= 24;
  const int s = blk % SS;
  const int b = blk / SS;
  const int lane = threadIdx.x;
  const int start = start_idx[0];
  const int j = lane * 2;                       // pair index base (0..62)

  if (slot >= 20) {                             // V: straight copy into cache
    const int hv = slot - 20;
    const float4 val = *(const float4*)(vin + (((size_t)(b * SS + s)) * HKV + hv) * DD + lane * 4);
    *(float4*)(present + ((((size_t)(b * 2 + 1)) * HKV + hv) * MAXP + start + s) * DD + lane * 4) = val;
    return;
  }
  const bool isq = (slot < 16);
  const int  h   = isq ? slot : slot - 16;
  const float* xin = isq ? (qin + (((size_t)(b * SS + s)) * HQ  + h) * DD)
                         : (kin + (((size_t)(b * SS + s)) * HKV + h) * DD);
  const float* w = isq ? qw : kw;

  float2 x1 = *(const float2*)(xin + j);
  float2 x2 = *(const float2*)(xin + j + 64);
  float ss2 = x1.x * x1.x + x1.y * x1.y + x2.x * x2.x + x2.y * x2.y;
#pragma unroll
  for (int off = 1; off < 32; off <<= 1) ss2 += __shfl_xor(ss2, off);
  const float inv = rsqrtf(ss2 * (1.0f / (float)DD) + 1e-6f);

  float2 w1 = *(const float2*)(w + j);
  float2 w2 = *(const float2*)(w + j + 64);
  x1.x *= inv * w1.x; x1.y *= inv * w1.y;
  x2.x *= inv * w2.x; x2.y *= inv * w2.y;

  const float* rp = rope + (size_t)(b * SS + s) * DD;
  float2 cs = *(const float2*)(rp + j);
  float2 sn = *(const float2*)(rp + j + 64);
  float2 o1, o2;
  o1.x = x1.x * cs.x - x2.x * sn.x;  o1.y = x1.y * cs.y - x2.y * sn.y;
  o2.x = x2.x * cs.x + x1.x * sn.x;  o2.y = x2.y * cs.y + x1.y * sn.y;

  if (isq) {
    float* qo = qout + (((size_t)(b * SS + s)) * HQ + h) * DD;
    *(float2*)(qo + j) = o1;  *(float2*)(qo + j + 64) = o2;
  } else {
    float* ko = present + ((((size_t)(b * 2)) * HKV + h) * MAXP + start + s) * DD;
    *(float2*)(ko + j) = o1;  *(float2*)(ko + j + 64) = o2;
  }
}

// =====================================================================
// Flash attention: one wave per (b, h, 16-query tile).
// scores = (Q Kt)/sqrt(D) with causal+ctx mask, online softmax, P@V.
// All matmuls via V_WMMA_F32_16X16X4_F32.
// =====================================================================
__global__ __launch_bounds__(32) void attn_fa(
    const float* __restrict__ qbuf, const float* __restrict__ present,
    float* __restrict__ obuf, const int* __restrict__ ctx_len,
    const int* __restrict__ start_idx) {
  __shared__ float Pl[16][17];

  const int lane = threadIdx.x;
  const int lm = lane & 15;
  const int hi = lane >> 4;

  int blk = blockIdx.x;
  const int qt = blk % (SS / 16); blk /= (SS / 16);
  const int h  = blk % HQ;
  const int b  = blk / HQ;
  const int hkv = h / GG;
  const int q0  = qt * 16;
  const int start = start_idx[0];
  const int ctx   = ctx_len[b];

  // Q tile as 32 A-fragments (K-chunks of 4 over D=128)
  v2f qa[32];
  const float* qrow = qbuf + (((size_t)(b * SS + q0 + lm)) * HQ + h) * DD;
#pragma unroll
  for (int kc = 0; kc < 32; ++kc) {
    const float2 t = *(const float2*)(qrow + kc * 4 + hi * 2);
    qa[kc].x = t.x; qa[kc].y = t.y;
  }

  const float* Kbase = present + (((size_t)(b * 2 + 0)) * HKV + hkv) * (size_t)MAXP * DD;
  const float* Vbase = present + (((size_t)(b * 2 + 1)) * HKV + hkv) * (size_t)MAXP * DD;

  v8f o[8];
#pragma unroll
  for (int t = 0; t < 8; ++t) o[t] = (v8f){0.f,0.f,0.f,0.f,0.f,0.f,0.f,0.f};
  float mrow[8], lrow[8];
#pragma unroll
  for (int r = 0; r < 8; ++r) { mrow[r] = -3.0e38f; lrow[r] = 0.0f; }

  const float scale = 0.088388347648318447f;   // 1/sqrt(128)
  int limit_max = start + q0 + 16;             // max(limit) over 16 rows
  if (ctx < limit_max) limit_max = ctx;
  const int ntiles = (limit_max + 15) >> 4;

  for (int kt = 0; kt < ntiles; ++kt) {
    const int p0 = kt * 16;

    // ---- scores: 16x16 tile via 32 WMMAs over D ----
    v8f c = {};
    const float* krow = Kbase + (size_t)(p0 + lm) * DD;
#pragma unroll
    for (int kc = 0; kc < 32; ++kc) {
      v2f bf;
      const float2 t = *(const float2*)(krow + kc * 4 + hi * 2);
      bf.x = t.x; bf.y = t.y;
      c = __builtin_amdgcn_wmma_f32_16x16x4_f32(false, qa[kc], false, bf, (short)0, c, false, false);
    }

    // ---- scale + mask (element: row = r+8*hi, key = p0+lm) ----
    const int p = p0 + lm;
#pragma unroll
    for (int r = 0; r < 8; ++r) {
      const int row  = r + 8 * hi;
      const int qpos = start + q0 + row;
      const bool ok  = (p <= qpos) && (p < ctx);
      c[r] = ok ? c[r] * scale : NEGV;
    }

    // ---- online softmax per row (16-lane half-wave reductions) ----
#pragma unroll
    for (int r = 0; r < 8; ++r) {
      float s = c[r];
      float rm = s;
#pragma unroll
      for (int off = 1; off < 16; off <<= 1) rm = fmaxf(rm, __shfl_xor(rm, off));
      const float mnew = fmaxf(mrow[r], rm);
      const float corr = __expf(mrow[r] - mnew);
      const float pv   = __expf(s - mnew);
      float rs = pv;
#pragma unroll
      for (int off = 1; off < 16; off <<= 1) rs += __shfl_xor(rs, off);
      lrow[r] = lrow[r] * corr + rs;
      mrow[r] = mnew;
#pragma unroll
      for (int t = 0; t < 8; ++t) o[t][r] *= corr;
      c[r] = pv;
    }

    // ---- D-layout -> A-layout through LDS ----
    __syncthreads();
#pragma unroll
    for (int r = 0; r < 8; ++r) Pl[r + 8 * hi][lm] = c[r];
    __syncthreads();
    v2f pa[4];
#pragma unroll
    for (int kc = 0; kc < 4; ++kc) {
      pa[kc].x = Pl[lm][kc * 4 + hi * 2 + 0];
      pa[kc].y = Pl[lm][kc * 4 + hi * 2 + 1];
    }

    // ---- P @ V : accumulate into 16x128 output (8 N-tiles x 4 K-chunks) ----
#pragma unroll
    for (int t = 0; t < 8; ++t) {
#pragma unroll
      for (int kc = 0; kc < 4; ++kc) {
        v2f bv;
        bv.x = Vbase[(size_t)(p0 + kc * 4 + hi * 2 + 0) * DD + t * 16 + lm];
        bv.y = Vbase[(size_t)(p0 + kc * 4 + hi * 2 + 1) * DD + t * 16 + lm];
        o[t] = __builtin_amdgcn_wmma_f32_16x16x4_f32(false, pa[kc], false, bv, (short)0, o[t], false, false);
      }
    }
  }

  // ---- epilogue: normalize and store [b, s, h, d] ----
#pragma unroll
  for (int r = 0; r < 8; ++r) {
    const int row = r + 8 * hi;
    const float inv = 1.0f / lrow[r];
    float* orow = obuf + (((size_t)(b * SS + q0 + row)) * HQ + h) * DD;
#pragma unroll
    for (int t = 0; t < 8; ++t) orow[t * 16 + lm] = o[t][r] * inv;
  }
}

// =====================================================================
// launch
// =====================================================================
extern "C" void kernel_launch(void* const* d_in, const int* in_sizes, int n_in,
                              void* d_out, int out_size, void* d_ws, size_t ws_size,
                              hipStream_t stream) {
  const float* hs    = (const float*)d_in[0];
  const float* past  = (const float*)d_in[1];
  const float* rope  = (const float*)d_in[2];
  const float* Wq    = (const float*)d_in[3];
  const float* Wk    = (const float*)d_in[4];
  const float* Wv    = (const float*)d_in[5];
  const float* Wo    = (const float*)d_in[6];
  const float* qw    = (const float*)d_in[7];
  const float* kw    = (const float*)d_in[8];
  const int*   ctx   = (const int*)d_in[9];
  const int*   start = (const int*)d_in[10];

  float* out      = (float*)d_out;
  float* attn_out = out;                               // B*S*HID
  float* present  = out + (size_t)BB * SS * HID;       // B*2*HKV*MAXP*D

  // workspace: qbuf (8MB) | kbuf (2MB) | vbuf (2MB) | abuf (8MB) = 20MB
  float* qbuf = (float*)d_ws;
  float* kbuf = qbuf + (size_t)BB * SS * HQ * DD;
  float* vbuf = kbuf + (size_t)BB * SS * HKV * DD;
  float* abuf = vbuf + (size_t)BB * SS * HKV * DD;

  const int M = BB * SS;                               // 1024

  // 1) QKV projections (f32 WMMA GEMMs)
  gemm_f32_wmma<<<dim3((HQ * DD) / 64,  M / 128), 256, 0, stream>>>(hs, Wq, qbuf, M, HQ * DD,  HID);
  gemm_f32_wmma<<<dim3((HKV * DD) / 64, M / 128), 256, 0, stream>>>(hs, Wk, kbuf, M, HKV * DD, HID);
  gemm_f32_wmma<<<dim3((HKV * DD) / 64, M / 128), 256, 0, stream>>>(hs, Wv, vbuf, M, HKV * DD, HID);

  // 2) present = past (then new slice overwritten by scatter)
  (void)hipMemcpyAsync(present, past,
                       (size_t)BB * 2 * HKV * MAXP * DD * sizeof(float),
                       hipMemcpyDeviceToDevice, stream);

  // 3) RMSNorm + RoPE + cache scatter
  normrope_scatter<<<BB * SS * (HQ + 2 * HKV), 32, 0, stream>>>(
      qbuf, kbuf, vbuf, rope, qw, kw, qbuf, present, start);

  // 4) flash attention over the cache
  attn_fa<<<BB * HQ * (SS / 16), 32, 0, stream>>>(qbuf, present, abuf, ctx, start);

  // 5) output projection
  gemm_f32_wmma<<<dim3(HID / 64, M / 128), 256, 0, stream>>>(abuf, Wo, attn_out, M, HID, HQ * DD);
}